// VectorQuantizer_47828755808923
// MI455X (gfx1250) — compile-verified
//
#include <hip/hip_runtime.h>

#define B_ROWS   65536
#define DIM      64
#define K_CODES  8192
#define TILE_N   128
#define LDS_STRIDE 68   // 64 + 4 pad: float4/b128 aligned, bank-conflict-free float2 reads
#define NT       (K_CODES / TILE_N)          // 64 tiles
#define TE       (TILE_N * LDS_STRIDE)       // floats per LDS tile buffer

// ---- output layout (float offsets, concatenated in reference return order) ----
#define OFF_ZQ     0
#define OFF_CODES  (B_ROWS * DIM)                 // 4194304
#define OFF_COMMIT (OFF_CODES + B_ROWS)           // 4259840
#define OFF_CBLOSS (OFF_COMMIT + 1)
#define OFF_VQ     (OFF_COMMIT + 2)
#define OFF_NEWCB  (OFF_COMMIT + 3)               // 4259843
#define OFF_NEWEC  (OFF_NEWCB + K_CODES * DIM)    // 4784131
#define OFF_NEWEW  (OFF_NEWEC + K_CODES)          // 4792323
#define OFF_USAGE  (OFF_NEWEW + K_CODES * DIM)    // 5316611

// ---- workspace layout (float offsets into d_ws) ----
#define WS_CSQ   0                                // float[8192]
#define WS_CNT   8192                             // float[8192]
#define WS_LOSS  16384                            // float[1]
#define WS_CODES 16388                            // int32[65536] (16B-aligned)
#define WS_ASUM  81924                            // float[524288] (16B-aligned)

typedef float v2f __attribute__((ext_vector_type(2)));
typedef float v8f __attribute__((ext_vector_type(8)));

#define EMA_A ((float)(0.99))
#define EMA_B ((float)(1.0 - 0.99))

// ---- CDNA5 async global->LDS path (ASYNCcnt), with graceful fallback ----
#ifndef __has_builtin
#define __has_builtin(x) 0
#endif
#if __has_builtin(__builtin_amdgcn_global_load_async_to_lds_b128)
#define HAVE_ASYNC_LDS 1
#else
#define HAVE_ASYNC_LDS 0
#endif

#if HAVE_ASYNC_LDS
// Builtin signature (from hipcc diagnostic): param0 = vector_size(16) int, addrspace(1);
// param1 = same vector type in LDS addrspace(3); then imm offset, imm cpol.
typedef int v4i_vs __attribute__((vector_size(16)));
typedef __attribute__((address_space(1))) v4i_vs as1_v4i;
typedef __attribute__((address_space(3))) v4i_vs as3_v4i;
#if __has_builtin(__builtin_amdgcn_s_wait_asynccnt)
#define WAIT_ASYNC(n) __builtin_amdgcn_s_wait_asynccnt(n)
#else
#define WAIT_ASYNC(n) asm volatile("s_wait_asynccnt %0" ::"i"(n) : "memory")
#endif
#else
#define WAIT_ASYNC(n)
#endif

// ---------------- Kernel 0: zero scratch + c_sq ----------------
__global__ __launch_bounds__(256) void vq_prep(const float* __restrict__ cb,
                                               float* __restrict__ ws) {
  int i = blockIdx.x * 256 + threadIdx.x;          // grid covers K*D
  if (i < K_CODES * DIM) ws[WS_ASUM + i] = 0.0f;
  if (i < K_CODES) {
    ws[WS_CNT + i] = 0.0f;
    const float* row = cb + i * DIM;
    float s = 0.0f;
    #pragma unroll 8
    for (int d = 0; d < DIM; ++d) s += row[d] * row[d];
    ws[WS_CSQ + i] = s;
  }
  if (i == 0) ws[WS_LOSS] = 0.0f;
}

// Stage one 128x64 f32 codebook tile into (padded) LDS.
// Async path: 8 x GLOBAL_LOAD_ASYNC_TO_LDS_B128 per thread, no VGPR roundtrip.
__device__ __forceinline__ void stage_tile(const float* __restrict__ tile_src,
                                           float* __restrict__ dst, int tid) {
  const float4* src4 = reinterpret_cast<const float4*>(tile_src);
  #pragma unroll
  for (int it = 0; it < 8; ++it) {
    int q  = tid + it * 256;        // 0..2047 float4 slots
    int c  = q >> 4;                // code within tile
    int c4 = q & 15;                // float4 column
#if HAVE_ASYNC_LDS
    __builtin_amdgcn_global_load_async_to_lds_b128(
        (as1_v4i*)(src4 + c * 16 + c4),
        (as3_v4i*)(dst + c * LDS_STRIDE + c4 * 4),
        0, 0);
#else
    float4 v = src4[c * 16 + c4];
    *reinterpret_cast<float4*>(dst + c * LDS_STRIDE + c4 * 4) = v;
#endif
  }
}

// ---------------- Kernel 1: WMMA distance + fused argmin ----------------
__global__ __launch_bounds__(256) void vq_argmin(const float* __restrict__ z,
                                                 const float* __restrict__ cb,
                                                 float* __restrict__ ws,
                                                 float* __restrict__ out_codes_f) {
  __shared__ __align__(16) float lds_cb[2 * TE];   // double-buffered tiles (~68KB)

  const int tid  = threadIdx.x;
  const int wave = tid >> 5;
  const int lane = tid & 31;
  const int l16  = lane & 15;
  const int g    = lane >> 4;          // half-wave group

  const int m0  = blockIdx.x * 128 + wave * 16;    // 16 rows per wave
  const int row = m0 + l16;

  // A fragments for this wave's 16x64 z tile (16 K-steps of 16x4 f32).
  // ISA layout: VGPR v, lane group g -> K = 4k + 2g + v  (consecutive float2)
  v2f a[16];
  #pragma unroll
  for (int k = 0; k < 16; ++k)
    a[k] = *reinterpret_cast<const v2f*>(z + row * DIM + 4 * k + 2 * g);

  float best[8];
  int   bidx[8];
  #pragma unroll
  for (int r = 0; r < 8; ++r) { best[r] = 3.4e38f; bidx[r] = 0; }

  const float* csq_g = ws + WS_CSQ;

  // prologue: kick off tile 0
  stage_tile(cb, lds_cb, tid);

  for (int t = 0; t < NT; ++t) {
    float* cur = lds_cb + (t & 1) * TE;
    float* nxt = lds_cb + ((t + 1) & 1) * TE;
    const int tile0 = t * TILE_N;

    __syncthreads();                       // all waves done reading `nxt` (tile t-1)
    if (t + 1 < NT) {
      stage_tile(cb + (tile0 + TILE_N) * DIM, nxt, tid);   // overlap with compute of t
      WAIT_ASYNC(8);                       // only tile t's 8 copies must have landed
    } else {
      WAIT_ASYNC(0);
    }
    __syncthreads();                       // tile t visible to all waves

    #pragma unroll
    for (int nb = 0; nb < 8; ++nb) {
      // B fragment: lane holds codebook[n][4k+2g .. +1] (B = C^T, 4x16 per step)
      const float* bp = &cur[(nb * 16 + l16) * LDS_STRIDE + 2 * g];
      v8f acc = {};
      #pragma unroll
      for (int k = 0; k < 16; ++k) {
        v2f b = *reinterpret_cast<const v2f*>(bp + 4 * k);
        acc = __builtin_amdgcn_wmma_f32_16x16x4_f32(
            false, a[k], false, b, (short)0, acc, false, false);
      }
      const int   n_global = tile0 + nb * 16 + l16;
      const float cs       = csq_g[n_global];   // 16 consecutive floats, L2-hot
      #pragma unroll
      for (int r = 0; r < 8; ++r) {
        // z_sq[m] is row-constant: argmin over (c_sq - 2*dot) is identical
        float dist = cs - 2.0f * acc[r];
        if (dist < best[r]) { best[r] = dist; bidx[r] = n_global; }  // ties keep lower n
      }
    }
  }

  // Cross-lane argmin within each 16-lane half (row m = r + 8*g spread over 16 lanes)
  #pragma unroll
  for (int r = 0; r < 8; ++r) {
    #pragma unroll
    for (int m = 8; m >= 1; m >>= 1) {
      float ov = __shfl_xor(best[r], m, 16);
      int   oi = __shfl_xor(bidx[r], m, 16);
      if (ov < best[r] || (ov == best[r] && oi < bidx[r])) { best[r] = ov; bidx[r] = oi; }
    }
  }
  if (l16 == 0) {
    int* codes_i = reinterpret_cast<int*>(ws + WS_CODES);
    #pragma unroll
    for (int r = 0; r < 8; ++r) {
      int m = m0 + r + 8 * g;
      codes_i[m] = bidx[r];
      out_codes_f[m] = (float)bidx[r];
    }
  }
}

// ---------------- Kernel 2: gather z_q, losses, segment scatter ----------------
__global__ __launch_bounds__(256) void vq_scatter(const float* __restrict__ z,
                                                  const float* __restrict__ cb,
                                                  float* __restrict__ ws,
                                                  float* __restrict__ out) {
  __shared__ float red[256];
  int i    = blockIdx.x * 256 + threadIdx.x;   // < B*D
  int rowi = i >> 6;
  int d    = i & 63;
  const int* codes_i = reinterpret_cast<const int*>(ws + WS_CODES);
  int   code = codes_i[rowi];
  float zv   = z[i];
  float q    = cb[code * DIM + d];
  out[OFF_ZQ + i] = q;                         // z_q_st forward value == z_q
  float diff = zv - q;

  atomicAdd(&ws[WS_ASUM + code * DIM + d], zv);
  if (d == 0) atomicAdd(&ws[WS_CNT + code], 1.0f);

  red[threadIdx.x] = diff * diff;
  __syncthreads();
  for (int s = 128; s >= 1; s >>= 1) {
    if (threadIdx.x < s) red[threadIdx.x] += red[threadIdx.x + s];
    __syncthreads();
  }
  if (threadIdx.x == 0) atomicAdd(&ws[WS_LOSS], red[0]);
}

// ---------------- Kernel 3: EMA update + losses ----------------
__global__ __launch_bounds__(256) void vq_update(const float* __restrict__ ema_count,
                                                 const float* __restrict__ ema_weight,
                                                 const float* __restrict__ usage,
                                                 const float* __restrict__ ws,
                                                 float* __restrict__ out) {
  int i = blockIdx.x * 256 + threadIdx.x;      // < K*D
  int k = i >> 6;
  int d = i & 63;
  float cntk    = ws[WS_CNT + k];
  float new_cnt = EMA_A * ema_count[k] + EMA_B * cntk;
  float new_w   = EMA_A * ema_weight[i] + EMA_B * ws[WS_ASUM + i];
  out[OFF_NEWEW + i] = new_w;
  out[OFF_NEWCB + i] = new_w / (new_cnt + 1e-5f);
  if (d == 0) {
    out[OFF_NEWEC + k] = new_cnt;
    out[OFF_USAGE + k] = usage[k] + cntk;
  }
  if (i == 0) {
    float S = ws[WS_LOSS];
    float c = S / (float)(B_ROWS * DIM);       // commitment == codebook loss value
    out[OFF_COMMIT] = c;
    out[OFF_CBLOSS] = c;
    out[OFF_VQ]     = c + 0.25f * c;
  }
}

extern "C" void kernel_launch(void* const* d_in, const int* in_sizes, int n_in,
                              void* d_out, int out_size, void* d_ws, size_t ws_size,
                              hipStream_t stream) {
  (void)in_sizes; (void)n_in; (void)out_size; (void)ws_size;
  const float* z          = (const float*)d_in[0];
  const float* cb         = (const float*)d_in[1];
  const float* ema_count  = (const float*)d_in[2];
  const float* ema_weight = (const float*)d_in[3];
  const float* usage      = (const float*)d_in[4];
  float* out = (float*)d_out;
  float* ws  = (float*)d_ws;

  vq_prep   <<<(K_CODES * DIM) / 256, 256, 0, stream>>>(cb, ws);
  vq_argmin <<<B_ROWS / 128,          256, 0, stream>>>(z, cb, ws, out + OFF_CODES);
  vq_scatter<<<(B_ROWS * DIM) / 256,  256, 0, stream>>>(z, cb, ws, out);
  vq_update <<<(K_CODES * DIM) / 256, 256, 0, stream>>>(ema_count, ema_weight, usage, ws, out);
}